// CRF_16338055594779
// MI455X (gfx1250) — compile-verified
//
#include <hip/hip_runtime.h>
#include <hip/hip_bf16.h>

typedef __attribute__((ext_vector_type(2))) float v2f;
typedef __attribute__((ext_vector_type(8))) float v8f;

#define LNUM 102          // n_labels (incl. start=100, stop=101)
#define LP   128          // padded label dim (8 tiles of 16)
#define ROWS 16           // batch rows per block
#define NEG_INF_F (-10000.0f)

// One block owns ROWS batch rows for the whole T loop.
// 256 threads = 8 waves; wave w computes output j-tile [16w, 16w+16).
__launch_bounds__(256, 1)
__global__ void crf_forward_wmma(const float* __restrict__ logits,   // [B, T, LNUM]
                                 const float* __restrict__ trans,    // [LNUM, LNUM]
                                 const int*   __restrict__ lens,     // [B]
                                 float*       __restrict__ out,      // [B]
                                 int T) {
    extern __shared__ float smem[];
    float* Et    = smem;                 // [LP][LP]  Et[k][j] = exp(trans[j,k]), zero pad
    float* alpha = Et    + LP * LP;      // [ROWS][LP] log-space state (pad = -1e4)
    float* xbuf  = alpha + ROWS * LP;    // [ROWS][LP] exp(alpha - Mrow)
    float* lg    = xbuf  + ROWS * LP;    // [ROWS][LP] staged logits for step t
    float* pm    = lg    + ROWS * LP;    // [ROWS][16] partial row maxes
    float* Mrow  = pm    + ROWS * 16;    // [ROWS]
    int*   lensh = (int*)(Mrow + ROWS);  // [ROWS]

    const int tid  = threadIdx.x;
    const int lane = tid & 31;
    const int wv   = tid >> 5;                 // wave id = j-tile
    const int b0   = blockIdx.x * ROWS;

    const int jcol = (wv << 4) + (lane & 15);  // output column this lane owns
    const int hk2  = (lane >> 4) << 1;         // K sub-offset per A/B fp32 layout
    const int mhi  = (lane >> 4) << 3;         // C/D row offset (0 or 8)
    const int rrow = tid >> 4;                 // phase-a/b row (0..15)
    const int seg  = tid & 15;                 // phase-a/b segment (8 elems each)

    // ---- one-time init: Et = exp(trans)^T (zero-padded), alpha0, lens ----
    for (int idx = tid; idx < LP * LP; idx += 256) {
        int k = idx >> 7, j = idx & (LP - 1);
        float v = 0.0f;
        if (k < LNUM && j < LNUM) v = __expf(trans[j * LNUM + k]);
        Et[idx] = v;
    }
    for (int idx = tid; idx < ROWS * LP; idx += 256) {
        int c = idx & (LP - 1);
        alpha[idx] = (c == LNUM - 2) ? 0.0f : NEG_INF_F;   // start label = L-2
    }
    if (tid < ROWS) lensh[tid] = lens[b0 + tid];
    __syncthreads();

    // ---- hoist all B-matrix WMMA fragments (constant over T) into registers ----
    // fp32 B frag (4x16): VGPR0 = K=k, VGPR1 = K=k+1 (lanes 16-31 use k+2,k+3)
    v2f bf[32];
#pragma unroll
    for (int kk = 0; kk < 32; ++kk) {
        int k = (kk << 2) + hk2;
        v2f b;
        b.x = Et[k * LP + jcol];
        b.y = Et[(k + 1) * LP + jcol];
        bf[kk] = b;
    }
    // per-lane sequence lengths for the 8 output rows this lane produces
    int myLen[8];
#pragma unroll
    for (int r = 0; r < 8; ++r) myLen[r] = lensh[r + mhi];

    // ---- sequential scan over time ----
    for (int t = 0; t < T; ++t) {
        // phase A: partial row maxes of alpha + stage this step's logits tile
        {
            const float* arow = alpha + rrow * LP + seg * 8;
            float mx = -1e30f;
#pragma unroll
            for (int i = 0; i < 8; ++i) mx = fmaxf(mx, arow[i]);
            pm[rrow * 16 + seg] = mx;
        }
        for (int idx = tid; idx < ROWS * LNUM; idx += 256) {
            int r = idx / LNUM, c = idx - r * LNUM;
            lg[r * LP + c] = logits[((size_t)(b0 + r) * T + t) * LNUM + c];
        }
        if (t + 1 < T && tid < 64) {  // prefetch next step's tile -> global_prefetch_b8
            int r = tid >> 2, c = (tid & 3) * 32;
            __builtin_prefetch(&logits[((size_t)(b0 + r) * T + (t + 1)) * LNUM + c], 0, 0);
        }
        __syncthreads();

        // phase B: reduce row maxes
        if (tid < ROWS) {
            float mx = pm[tid * 16];
#pragma unroll
            for (int i = 1; i < 16; ++i) mx = fmaxf(mx, pm[tid * 16 + i]);
            Mrow[tid] = mx;
        }
        __syncthreads();

        // phase B2: x = exp(alpha - Mrow)  (pads give exp(-1e4 - M) == 0)
        {
            const float mr = Mrow[rrow];
            const float* arow = alpha + rrow * LP + seg * 8;
            float*       xrow = xbuf  + rrow * LP + seg * 8;
#pragma unroll
            for (int i = 0; i < 8; ++i) xrow[i] = __expf(arow[i] - mr);
        }
        __syncthreads();

        // phase C: fp32 WMMA chain  acc[16x16] += A[16x4] * B[4x16]
        v8f acc = {};
#pragma unroll
        for (int kk = 0; kk < 32; ++kk) {
            const v2f a = *(const v2f*)&xbuf[(lane & 15) * LP + (kk << 2) + hk2];
            acc = __builtin_amdgcn_wmma_f32_16x16x4_f32(
                false, a, false, bf[kk], (short)0, acc, false, false);
        }
        // alpha'[m,j] = logit[m,t,j] + log(acc) + Mrow[m], masked by t < len[m]
        if (jcol < LNUM) {
#pragma unroll
            for (int r = 0; r < 8; ++r) {
                const int m = r + mhi;
                if (t < myLen[r]) {
                    alpha[m * LP + jcol] = lg[m * LP + jcol] + __logf(acc[r]) + Mrow[m];
                }
            }
        }
        __syncthreads();
    }

    // ---- finalize: alpha += trans[stop,:]; out = logsumexp(alpha) ----
    if (tid < ROWS) {
        const float* arow = alpha + tid * LP;
        const float* tr   = trans + (LNUM - 1) * LNUM;   // stop label row
        float mx = -1e30f;
        for (int k = 0; k < LNUM; ++k) mx = fmaxf(mx, arow[k] + tr[k]);
        float s = 0.0f;
        for (int k = 0; k < LNUM; ++k) s += __expf(arow[k] + tr[k] - mx);
        out[b0 + tid] = __logf(s) + mx;
    }
}

extern "C" void kernel_launch(void* const* d_in, const int* in_sizes, int n_in,
                              void* d_out, int out_size, void* d_ws, size_t ws_size,
                              hipStream_t stream) {
    const float* logits = (const float*)d_in[0];   // [B, T, L] f32
    const float* trans  = (const float*)d_in[1];   // [L, L]    f32
    const int*   lens   = (const int*)d_in[2];     // [B]       i32
    float*       out    = (float*)d_out;           // [B]       f32

    const int B = in_sizes[2];
    const int T = (int)((long long)in_sizes[0] / ((long long)B * LNUM));

    const size_t shmem =
        (size_t)(LP * LP + 3 * ROWS * LP + ROWS * 16 + ROWS) * sizeof(float) +
        ROWS * sizeof(int);

    crf_forward_wmma<<<B / ROWS, 256, shmem, stream>>>(logits, trans, lens, out, T);
}